// Decoder_29489245454440
// MI455X (gfx1250) — compile-verified
//
#include <hip/hip_runtime.h>
#include <hip/hip_bf16.h>

// Problem constants (match reference)
#define BB 128
#define NN 2048
#define DD 512
#define SS 64

typedef __attribute__((ext_vector_type(2))) float v2f;
typedef __attribute__((ext_vector_type(8))) float v8f;

__device__ __forceinline__ float neg_inf_f() { return __uint_as_float(0xFF800000u); }

// ---------------------------------------------------------------------------
// WMMA fp32 GEMM:  Dst[M,Nc] = A[M,K] * W[Nc,K]^T + bias[Nc]
// (x @ W.T + b, W stored row-major [out,in] as in the reference)
// One wave computes one 16x16 tile using V_WMMA_F32_16X16X4_F32.
//  A 16x4 layout : lane m = lane&15, VGPR0 holds K = 2*(lane>>4), VGPR1 = +1
//  B 4x16 layout : lane n = lane&15, same K striping (B row striped over lanes)
//  C/D 16x16     : VGPR v -> row = v + 8*(lane>>4), col = lane&15
// ---------------------------------------------------------------------------
__global__ __launch_bounds__(256) void wmma_xwt_bias_kernel(
    const float* __restrict__ A, const float* __restrict__ W,
    const float* __restrict__ bias, float* __restrict__ Dst,
    int M, int K, int Nc) {
  const int lane   = threadIdx.x & 31;
  const int waveG  = (blockIdx.x * blockDim.x + threadIdx.x) >> 5;
  const int ntiles = Nc >> 4;
  const int tileN  = waveG % ntiles;
  const int tileM  = waveG / ntiles;
  if (tileM * 16 >= M) return;  // wave-uniform

  const int mrow = tileM * 16 + (lane & 15);
  const int ncol = tileN * 16 + (lane & 15);
  const int kb   = (lane >> 4) << 1;  // 0 for lanes 0-15, 2 for lanes 16-31

  const float* __restrict__ arow = A + (size_t)mrow * K;
  const float* __restrict__ wrow = W + (size_t)ncol * K;

  v8f c = {};
  for (int k0 = 0; k0 < K; k0 += 4) {
    v2f a, b;
    a.x = arow[k0 + kb];
    a.y = arow[k0 + kb + 1];
    b.x = wrow[k0 + kb];
    b.y = wrow[k0 + kb + 1];
    // (neg_a, A, neg_b, B, c_mod, C, reuse_a, reuse_b)
    c = __builtin_amdgcn_wmma_f32_16x16x4_f32(false, a, false, b, (short)0, c,
                                              false, false);
  }

  const float bv = bias[ncol];
  const int rbase = tileM * 16 + ((lane >> 4) << 3);
#pragma unroll
  for (int v = 0; v < 8; ++v) {
    Dst[(size_t)(rbase + v) * Nc + ncol] = c[v] + bv;
  }
}

// ---------------------------------------------------------------------------
// Fused single-query attention with online softmax (flash-style).
// One workgroup (8 waves) per batch. Each wave streams node rows (whole wave
// per row: 512 floats = 16/lane), computes the dot with q via a wave32
// butterfly reduce, and maintains running (max, sum, context[16]/lane).
// Masked rows are skipped entirely (weight 0) -> ~30% bandwidth saved.
// Cross-wave combine through LDS with standard log-sum-exp rescaling.
// ---------------------------------------------------------------------------
__global__ __launch_bounds__(256) void attn_ctx_kernel(
    const float* __restrict__ E, const float* __restrict__ Q,
    const unsigned char* __restrict__ mask, float* __restrict__ Ctx) {
  __shared__ float q_s[DD];
  __shared__ float ctx_s[8][DD];
  __shared__ float m_s[8];
  __shared__ float l_s[8];

  const int b   = blockIdx.x;
  const int tid = threadIdx.x;
  for (int i = tid; i < DD; i += 256) q_s[i] = Q[(size_t)b * DD + i];
  __syncthreads();

  const int wave = tid >> 5;
  const int lane = tid & 31;
  const float scale = 0.044194173824159216f;  // 1/sqrt(512)

  float qreg[16];
#pragma unroll
  for (int j = 0; j < 16; ++j) qreg[j] = q_s[lane * 16 + j];

  float m = neg_inf_f();
  float l = 0.0f;
  float ctx[16];
#pragma unroll
  for (int j = 0; j < 16; ++j) ctx[j] = 0.0f;

  const float* __restrict__ Eb = E + (size_t)b * NN * DD;
  const unsigned char* __restrict__ mb = mask + (size_t)b * NN;

  for (int n = wave; n < NN; n += 8) {
    if (mb[n]) continue;  // wave-uniform skip: no row load for masked nodes
    const float4* __restrict__ row =
        reinterpret_cast<const float4*>(Eb + (size_t)n * DD + lane * 16);
    float e[16];
    float4 t0 = row[0], t1 = row[1], t2 = row[2], t3 = row[3];
    e[0] = t0.x; e[1] = t0.y; e[2]  = t0.z; e[3]  = t0.w;
    e[4] = t1.x; e[5] = t1.y; e[6]  = t1.z; e[7]  = t1.w;
    e[8] = t2.x; e[9] = t2.y; e[10] = t2.z; e[11] = t2.w;
    e[12] = t3.x; e[13] = t3.y; e[14] = t3.z; e[15] = t3.w;

    float p = 0.0f;
#pragma unroll
    for (int j = 0; j < 16; ++j) p = fmaf(e[j], qreg[j], p);
#pragma unroll
    for (int off = 16; off > 0; off >>= 1) p += __shfl_xor(p, off, 32);

    const float s  = p * scale;
    const float mn = fmaxf(m, s);
    const float r  = __expf(m - mn);  // first iter: exp(-inf) = 0
    const float w  = __expf(s - mn);
    l = l * r + w;
#pragma unroll
    for (int j = 0; j < 16; ++j) ctx[j] = ctx[j] * r + w * e[j];
    m = mn;
  }

#pragma unroll
  for (int j = 0; j < 16; ++j) ctx_s[wave][lane * 16 + j] = ctx[j];
  if (lane == 0) { m_s[wave] = m; l_s[wave] = l; }
  __syncthreads();

  float gm = neg_inf_f();
#pragma unroll
  for (int w8 = 0; w8 < 8; ++w8) gm = fmaxf(gm, m_s[w8]);
  float f[8];
  float gl = 0.0f;
#pragma unroll
  for (int w8 = 0; w8 < 8; ++w8) {
    f[w8] = __expf(m_s[w8] - gm);
    gl += f[w8] * l_s[w8];
  }
  const float inv = 1.0f / gl;
  for (int d = tid; d < DD; d += 256) {
    float acc = 0.0f;
#pragma unroll
    for (int w8 = 0; w8 < 8; ++w8) acc += f[w8] * ctx_s[w8][d];
    Ctx[(size_t)b * DD + d] = acc * inv;
  }
}

// ---------------------------------------------------------------------------
// logits[b,n] = scale * dot(E[b,n,:], scores[b,:]) ; masked -> -inf
// Block = 8 waves, handles 64 consecutive node rows of one batch.
// scores[b] cached in LDS; whole-wave row dot + butterfly reduce.
// ---------------------------------------------------------------------------
__global__ __launch_bounds__(256) void logits_kernel(
    const float* __restrict__ E, const float* __restrict__ Sc,
    const unsigned char* __restrict__ mask, float* __restrict__ out) {
  __shared__ float s_s[DD];
  const int b     = blockIdx.x >> 5;   // 32 chunks per batch
  const int chunk = blockIdx.x & 31;   // 64 rows per chunk
  const int tid   = threadIdx.x;
  for (int i = tid; i < DD; i += 256) s_s[i] = Sc[(size_t)b * DD + i];
  __syncthreads();

  const int wave = tid >> 5;
  const int lane = tid & 31;
  const float scale = 0.044194173824159216f;

  float sreg[16];
#pragma unroll
  for (int j = 0; j < 16; ++j) sreg[j] = s_s[lane * 16 + j];

  const float* __restrict__ Eb = E + (size_t)b * NN * DD;
  const unsigned char* __restrict__ mb = mask + (size_t)b * NN;
  float* __restrict__ ob = out + (size_t)b * NN;
  const int n0 = chunk * 64;

  for (int i = 0; i < 8; ++i) {
    const int n = n0 + i * 8 + wave;  // 8 waves touch 8 consecutive rows
    if (mb[n]) {
      if (lane == 0) ob[n] = neg_inf_f();
      continue;
    }
    const float4* __restrict__ row =
        reinterpret_cast<const float4*>(Eb + (size_t)n * DD + lane * 16);
    float4 t0 = row[0], t1 = row[1], t2 = row[2], t3 = row[3];
    float p = 0.0f;
    p = fmaf(t0.x, sreg[0], p);  p = fmaf(t0.y, sreg[1], p);
    p = fmaf(t0.z, sreg[2], p);  p = fmaf(t0.w, sreg[3], p);
    p = fmaf(t1.x, sreg[4], p);  p = fmaf(t1.y, sreg[5], p);
    p = fmaf(t1.z, sreg[6], p);  p = fmaf(t1.w, sreg[7], p);
    p = fmaf(t2.x, sreg[8], p);  p = fmaf(t2.y, sreg[9], p);
    p = fmaf(t2.z, sreg[10], p); p = fmaf(t2.w, sreg[11], p);
    p = fmaf(t3.x, sreg[12], p); p = fmaf(t3.y, sreg[13], p);
    p = fmaf(t3.z, sreg[14], p); p = fmaf(t3.w, sreg[15], p);
#pragma unroll
    for (int off = 16; off > 0; off >>= 1) p += __shfl_xor(p, off, 32);
    if (lane == 0) ob[n] = p * scale;
  }
}

// ---------------------------------------------------------------------------
extern "C" void kernel_launch(void* const* d_in, const int* in_sizes, int n_in,
                              void* d_out, int out_size, void* d_ws,
                              size_t ws_size, hipStream_t stream) {
  const float* E  = (const float*)d_in[0];          // [B,N,D]
  const float* TS = (const float*)d_in[1];          // [B,S]
  const unsigned char* mask = (const unsigned char*)d_in[2];  // [B,N] bool
  const float* Wq = (const float*)d_in[3];          // [D,S]
  const float* bq = (const float*)d_in[4];          // [D]
  const float* Ws = (const float*)d_in[5];          // [D,D]
  const float* bs = (const float*)d_in[6];          // [D]
  float* out = (float*)d_out;                       // [B,N]

  float* Q   = (float*)d_ws;          // [B,D]
  float* Ctx = Q + (size_t)BB * DD;   // [B,D]
  float* Sc  = Ctx + (size_t)BB * DD; // [B,D]

  // 1) Q = TS @ Wq^T + bq   (M=128, K=64, N=512) -> 256 tiles / 8 waves
  {
    int tiles = (BB / 16) * (DD / 16);
    wmma_xwt_bias_kernel<<<tiles / 8, 256, 0, stream>>>(TS, Wq, bq, Q, BB, SS, DD);
  }
  // 2) fused masked-softmax attention + context (one WG per batch)
  attn_ctx_kernel<<<BB, 256, 0, stream>>>(E, Q, mask, Ctx);
  // 3) Sc = Ctx @ Ws^T + bs  (M=128, K=512, N=512)
  {
    int tiles = (BB / 16) * (DD / 16);
    wmma_xwt_bias_kernel<<<tiles / 8, 256, 0, stream>>>(Ctx, Ws, bs, Sc, BB, DD, DD);
  }
  // 4) logits = scale * E . Sc, masked -> -inf
  logits_kernel<<<BB * 32, 256, 0, stream>>>(E, Sc, mask, out);
}